// AttentionLayer_22299470201187
// MI455X (gfx1250) — compile-verified
//
#include <hip/hip_runtime.h>
#include <hip/hip_bf16.h>

// Problem dims (from setup_inputs): B=64, P=512, Hy(Q)=512, H=1024, fp32.
static constexpr int Bd = 64;
static constexpr int Pd = 512;
static constexpr int Qd = 512;   // Hy
static constexpr int Hd = 1024;
static constexpr float NEGV = -1.0e9f;

typedef __attribute__((ext_vector_type(2))) float v2f;
typedef __attribute__((ext_vector_type(8))) float v8f;

__device__ __forceinline__ v8f wmma_f32(v2f a, v2f b, v8f c) {
  // V_WMMA_F32_16X16X4_F32 : D = A(16x4,f32) x B(4x16,f32) + C(16x16,f32)
  return __builtin_amdgcn_wmma_f32_16x16x4_f32(
      /*neg_a=*/false, a, /*neg_b=*/false, b,
      /*c_mod=*/(short)0, c, /*reuse_a=*/false, /*reuse_b=*/false);
}

// -------------------------------------------------------------------------
// Kernel 1: scores[b,p,q] = sum_h premise[b,p,h]*hypothesis[b,q,h], masked.
// One wave -> 16(p) x 64(q) tile; 8 waves/block cover all Q=512 for a p-tile.
// A frag: lane m=(l&15) holds premise row p0+m, K pair at k+2*(l>>4)  (float2)
// B frag: lane n=(l&15) holds hypothesis row q0+n (== B^T column), same K pair
// -------------------------------------------------------------------------
__global__ __launch_bounds__(256) void scores_kernel(
    const float* __restrict__ prem, const float* __restrict__ hyp,
    const int* __restrict__ pmask, const int* __restrict__ hmask,
    float* __restrict__ S) {
  const int b    = blockIdx.x;
  const int p0   = blockIdx.y * 16;
  const int wave = threadIdx.x >> 5;
  const int lane = threadIdx.x & 31;
  const int half = lane >> 4;
  const int m    = lane & 15;
  const int q0   = wave * 64;

  const float* arow = prem + (size_t)(b * Pd + p0 + m) * Hd + 2 * half;
  const float* brow = hyp  + (size_t)(b * Qd + q0 + m) * Hd + 2 * half;

  v8f acc0 = {}, acc1 = {}, acc2 = {}, acc3 = {};
  for (int k = 0; k < Hd; k += 4) {
    v2f a  = *(const v2f*)(arow + k);
    v2f b0 = *(const v2f*)(brow + k);
    v2f b1 = *(const v2f*)(brow + (size_t)16 * Hd + k);
    v2f b2 = *(const v2f*)(brow + (size_t)32 * Hd + k);
    v2f b3 = *(const v2f*)(brow + (size_t)48 * Hd + k);
    acc0 = wmma_f32(a, b0, acc0);
    acc1 = wmma_f32(a, b1, acc1);
    acc2 = wmma_f32(a, b2, acc2);
    acc3 = wmma_f32(a, b3, acc3);
  }

  int pok[8];
#pragma unroll
  for (int r = 0; r < 8; ++r) pok[r] = pmask[b * Pd + p0 + r + 8 * half];

  v8f accs[4] = {acc0, acc1, acc2, acc3};
#pragma unroll
  for (int j = 0; j < 4; ++j) {
    const int q  = q0 + j * 16 + m;
    const int hv = hmask[b * Qd + q];
#pragma unroll
    for (int r = 0; r < 8; ++r) {
      const int p = p0 + r + 8 * half;
      const float v = (hv && pok[r]) ? accs[j][r] : NEGV;
      S[(size_t)(b * Pd + p) * Qd + q] = v;
    }
  }
}

// -------------------------------------------------------------------------
// Kernel 2: per-row (over q) max and sum(exp) for p2h softmax.
// -------------------------------------------------------------------------
__global__ __launch_bounds__(128) void row_stats_kernel(
    const float* __restrict__ S, float* __restrict__ rMax, float* __restrict__ rSum) {
  __shared__ float red[128];
  const int row = blockIdx.x;          // b*P + p
  const int tid = threadIdx.x;
  const float* r = S + (size_t)row * Qd;

  float m = -3.0e38f;
  for (int q = tid; q < Qd; q += 128) m = fmaxf(m, r[q]);
  red[tid] = m; __syncthreads();
  for (int s = 64; s > 0; s >>= 1) {
    if (tid < s) red[tid] = fmaxf(red[tid], red[tid + s]);
    __syncthreads();
  }
  const float bm = red[0]; __syncthreads();

  float sum = 0.f;
  for (int q = tid; q < Qd; q += 128) sum += __expf(r[q] - bm);
  red[tid] = sum; __syncthreads();
  for (int s = 64; s > 0; s >>= 1) {
    if (tid < s) red[tid] += red[tid + s];
    __syncthreads();
  }
  if (tid == 0) { rMax[row] = bm; rSum[row] = red[0]; }
}

// -------------------------------------------------------------------------
// Kernel 3: per-column (over p) max and sum(exp) for h2p softmax.
// block = 16(q) x 16(p-lanes); reads are coalesced across the 16 q's.
// -------------------------------------------------------------------------
__global__ __launch_bounds__(256) void col_stats_kernel(
    const float* __restrict__ S, float* __restrict__ cMax, float* __restrict__ cSum) {
  __shared__ float red[16][17];
  const int b  = blockIdx.x;
  const int q  = blockIdx.y * 16 + threadIdx.x;
  const int ty = threadIdx.y;
  const float* col = S + (size_t)(b * Pd) * Qd + q;

  float m = -3.0e38f;
  for (int p = ty; p < Pd; p += 16) m = fmaxf(m, col[(size_t)p * Qd]);
  red[ty][threadIdx.x] = m; __syncthreads();
  for (int s = 8; s > 0; s >>= 1) {
    if (ty < s) red[ty][threadIdx.x] = fmaxf(red[ty][threadIdx.x], red[ty + s][threadIdx.x]);
    __syncthreads();
  }
  const float bm = red[0][threadIdx.x]; __syncthreads();

  float sum = 0.f;
  for (int p = ty; p < Pd; p += 16) sum += __expf(col[(size_t)p * Qd] - bm);
  red[ty][threadIdx.x] = sum; __syncthreads();
  for (int s = 8; s > 0; s >>= 1) {
    if (ty < s) red[ty][threadIdx.x] += red[ty + s][threadIdx.x];
    __syncthreads();
  }
  if (ty == 0) { cMax[b * Qd + q] = bm; cSum[b * Qd + q] = red[0][threadIdx.x]; }
}

// -------------------------------------------------------------------------
// Kernel 4: attended_hypothesis[b,p,h] = softmax_q(scores) @ hypothesis.
// A fragment built on the fly: exp(score - rowMax); 1/rowSum folded at writeout.
// -------------------------------------------------------------------------
__global__ __launch_bounds__(256) void att_hyp_kernel(
    const float* __restrict__ S, const float* __restrict__ hyp,
    const float* __restrict__ rMax, const float* __restrict__ rSum,
    float* __restrict__ O) {
  const int b    = blockIdx.x;
  const int p0   = blockIdx.y * 16;
  const int wave = threadIdx.x >> 5;
  const int lane = threadIdx.x & 31;
  const int half = lane >> 4;
  const int m    = lane & 15;
  const int h0   = (blockIdx.z * 8 + wave) * 64;

  const int prow   = b * Pd + p0 + m;
  const float rm   = rMax[prow];
  const float* srow = S + (size_t)prow * Qd + 2 * half;
  const float* bcol = hyp + (size_t)(b * Qd) * Hd + h0 + m;

  v8f acc0 = {}, acc1 = {}, acc2 = {}, acc3 = {};
  for (int k = 0; k < Qd; k += 4) {
    v2f s = *(const v2f*)(srow + k);
    v2f a; a.x = __expf(s.x - rm); a.y = __expf(s.y - rm);
    const float* bk = bcol + (size_t)(k + 2 * half) * Hd;
    v2f b0, b1, b2, b3;
    b0.x = bk[0];  b0.y = bk[Hd + 0];
    b1.x = bk[16]; b1.y = bk[Hd + 16];
    b2.x = bk[32]; b2.y = bk[Hd + 32];
    b3.x = bk[48]; b3.y = bk[Hd + 48];
    acc0 = wmma_f32(a, b0, acc0);
    acc1 = wmma_f32(a, b1, acc1);
    acc2 = wmma_f32(a, b2, acc2);
    acc3 = wmma_f32(a, b3, acc3);
  }

  v8f accs[4] = {acc0, acc1, acc2, acc3};
#pragma unroll
  for (int r = 0; r < 8; ++r) {
    const int p = p0 + r + 8 * half;
    const float inv = 1.0f / rSum[b * Pd + p];
#pragma unroll
    for (int j = 0; j < 4; ++j)
      O[(size_t)(b * Pd + p) * Hd + h0 + j * 16 + m] = accs[j][r] * inv;
  }
}

// -------------------------------------------------------------------------
// Kernel 5: attended_premise[b,q,h] = softmax_p(scores^T) @ premise.
// A fragment: exp(scores[b,k,q] - colMax) (transposed score reads).
// -------------------------------------------------------------------------
__global__ __launch_bounds__(256) void att_prem_kernel(
    const float* __restrict__ S, const float* __restrict__ prem,
    const float* __restrict__ cMax, const float* __restrict__ cSum,
    float* __restrict__ O) {
  const int b    = blockIdx.x;
  const int q0   = blockIdx.y * 16;
  const int wave = threadIdx.x >> 5;
  const int lane = threadIdx.x & 31;
  const int half = lane >> 4;
  const int m    = lane & 15;
  const int h0   = (blockIdx.z * 8 + wave) * 64;

  const int qrow   = b * Qd + q0 + m;
  const float cm   = cMax[qrow];
  const float* scol = S + (size_t)(b * Pd) * Qd + q0 + m;   // + k*Qd
  const float* brow = prem + (size_t)(b * Pd) * Hd + h0 + m; // + k*Hd

  v8f acc0 = {}, acc1 = {}, acc2 = {}, acc3 = {};
  for (int k = 0; k < Pd; k += 4) {
    const int kk = k + 2 * half;
    v2f a;
    a.x = __expf(scol[(size_t)kk * Qd] - cm);
    a.y = __expf(scol[(size_t)(kk + 1) * Qd] - cm);
    const float* bk = brow + (size_t)kk * Hd;
    v2f b0, b1, b2, b3;
    b0.x = bk[0];  b0.y = bk[Hd + 0];
    b1.x = bk[16]; b1.y = bk[Hd + 16];
    b2.x = bk[32]; b2.y = bk[Hd + 32];
    b3.x = bk[48]; b3.y = bk[Hd + 48];
    acc0 = wmma_f32(a, b0, acc0);
    acc1 = wmma_f32(a, b1, acc1);
    acc2 = wmma_f32(a, b2, acc2);
    acc3 = wmma_f32(a, b3, acc3);
  }

  v8f accs[4] = {acc0, acc1, acc2, acc3};
#pragma unroll
  for (int r = 0; r < 8; ++r) {
    const int q = q0 + r + 8 * half;
    const float inv = 1.0f / cSum[b * Qd + q];
#pragma unroll
    for (int j = 0; j < 4; ++j)
      O[(size_t)(b * Qd + q) * Hd + h0 + j * 16 + m] = accs[j][r] * inv;
  }
}

// -------------------------------------------------------------------------
// Kernel 6: finalize p2h in place (scores -> softmax rows). Runs last.
// -------------------------------------------------------------------------
__global__ __launch_bounds__(128) void p2h_finalize_kernel(
    float* __restrict__ S, const float* __restrict__ rMax, const float* __restrict__ rSum) {
  const int row = blockIdx.x;
  const float rm = rMax[row];
  const float inv = 1.0f / rSum[row];
  float* r = S + (size_t)row * Qd;
  for (int q = threadIdx.x; q < Qd; q += 128)
    r[q] = __expf(r[q] - rm) * inv;
}

extern "C" void kernel_launch(void* const* d_in, const int* in_sizes, int n_in,
                              void* d_out, int out_size, void* d_ws, size_t ws_size,
                              hipStream_t stream) {
  (void)in_sizes; (void)n_in; (void)out_size; (void)ws_size;

  const float* premise    = (const float*)d_in[0];
  const float* hypothesis = (const float*)d_in[1];
  const int*   pmask      = (const int*)d_in[2];
  const int*   hmask      = (const int*)d_in[3];

  float* out = (float*)d_out;
  float* att_prem = out;                                       // B*Q*H
  float* att_hyp  = out + (size_t)Bd * Qd * Hd;                // B*P*H
  float* scores   = att_hyp + (size_t)Bd * Pd * Hd;            // B*P*Q (becomes p2h)

  float* wsf  = (float*)d_ws;                                  // 512 KB of stats
  float* rMax = wsf;
  float* rSum = wsf + (size_t)Bd * Pd;
  float* cMax = wsf + (size_t)2 * Bd * Pd;
  float* cSum = wsf + (size_t)2 * Bd * Pd + (size_t)Bd * Qd;

  // 1) scores = premise @ hypothesis^T (masked), into p2h slot
  scores_kernel<<<dim3(Bd, Pd / 16), 256, 0, stream>>>(
      premise, hypothesis, pmask, hmask, scores);

  // 2) softmax statistics
  row_stats_kernel<<<Bd * Pd, 128, 0, stream>>>(scores, rMax, rSum);
  col_stats_kernel<<<dim3(Bd, Qd / 16), dim3(16, 16), 0, stream>>>(scores, cMax, cSum);

  // 3) attended outputs (read raw scores + stats)
  att_hyp_kernel<<<dim3(Bd, Pd / 16, Hd / (64 * 8)), 256, 0, stream>>>(
      scores, hypothesis, rMax, rSum, att_hyp);
  att_prem_kernel<<<dim3(Bd, Qd / 16, Hd / (64 * 8)), 256, 0, stream>>>(
      scores, premise, cMax, cSum, att_prem);

  // 4) turn scores into p2h in place (after all raw-score consumers)
  p2h_finalize_kernel<<<Bd * Pd, 128, 0, stream>>>(scores, rMax, rSum);
}